// GNNModel_74560632259284
// MI455X (gfx1250) — compile-verified
//
#include <hip/hip_runtime.h>
#include <hip/hip_bf16.h>
#include <math.h>

typedef float v2f __attribute__((ext_vector_type(2)));
typedef float v8f __attribute__((ext_vector_type(8)));

#define HDIM 128

// ---------------- degree / dinv ----------------
__global__ void k_init_deg(float* deg, int n) {
    int i = blockIdx.x * blockDim.x + threadIdx.x;
    if (i < n) deg[i] = 1.0f;                     // self-loop weight
}

__global__ void k_deg_scatter(const int* __restrict__ dst, float* deg, int e) {
    int i = blockIdx.x * blockDim.x + threadIdx.x;
    if (i < e) atomicAdd(&deg[dst[i]], 1.0f);
}

__global__ void k_deg_rsqrt(float* deg, int n) {
    int i = blockIdx.x * blockDim.x + threadIdx.x;
    if (i < n) deg[i] = rsqrtf(deg[i]);
}

// ---------------- WMMA GEMM: T = A @ W ; G = T * dinv^2 ----------------
// A: [N,128] row-major, W: [128,128] row-major.
// One 16x16 C tile per wave; 8 waves/block -> one 16-row x 128-col block.
__global__ __launch_bounds__(256)
void k_gemm_wmma(const float* __restrict__ A, const float* __restrict__ W,
                 float* __restrict__ T, float* __restrict__ G,
                 const float* __restrict__ dinv) {
    const int wave = threadIdx.x >> 5;
    const int lane = threadIdx.x & 31;
    const int half = lane >> 4;          // 0: K 0..1, 1: K 2..3
    const int lrow = lane & 15;          // M index for A, N index for B/C/D
    const int rowbase = blockIdx.x * 16;
    const int colbase = wave * 16;

    const float* arow = A + (size_t)(rowbase + lrow) * HDIM + 2 * half;
    v8f c = {};
    #pragma unroll 4
    for (int k0 = 0; k0 < HDIM; k0 += 4) {
        v2f a; a.x = arow[k0]; a.y = arow[k0 + 1];
        const int kk = k0 + 2 * half;
        v2f b; b.x = W[kk * HDIM + colbase + lrow];
               b.y = W[(kk + 1) * HDIM + colbase + lrow];
        c = __builtin_amdgcn_wmma_f32_16x16x4_f32(false, a, false, b,
                                                  (short)0, c, false, false);
    }
    #pragma unroll
    for (int i = 0; i < 8; ++i) {
        const int row = rowbase + i + 8 * half;   // C/D: vgpr i -> rows i, i+8
        const int col = colbase + lrow;
        const float v  = c[i];
        const float di = dinv[row];
        T[(size_t)row * HDIM + col] = v;
        G[(size_t)row * HDIM + col] = v * di * di;   // self-loop init
    }
}

// ---------------- edge scatter: G[dst] += T[src] * dinv[src]*dinv[dst] ----------------
__global__ __launch_bounds__(256)
void k_edge_scatter(const float* __restrict__ T, const int* __restrict__ src,
                    const int* __restrict__ dst, const float* __restrict__ dinv,
                    float* __restrict__ G, int e) {
    int idx = blockIdx.x * blockDim.x + threadIdx.x;
    if (idx >= e * 32) return;
    const int eg = idx >> 5;              // edge (wave-uniform: 32 threads/edge)
    const int ch = (idx & 31) * 4;        // 4-channel group
    const int s = src[eg], d = dst[eg];
    const float nrm = dinv[s] * dinv[d];
    const float4 v = *reinterpret_cast<const float4*>(T + (size_t)s * HDIM + ch);
    float* gp = G + (size_t)d * HDIM + ch;
    atomicAdd(gp + 0, v.x * nrm);
    atomicAdd(gp + 1, v.y * nrm);
    atomicAdd(gp + 2, v.z * nrm);
    atomicAdd(gp + 3, v.w * nrm);
}

// ---------------- H = relu(G + bias[col]) ----------------
__global__ __launch_bounds__(256)
void k_bias_relu(const float* __restrict__ G, const float* __restrict__ bias,
                 float* __restrict__ H, int total4) {
    int i = blockIdx.x * blockDim.x + threadIdx.x;
    if (i >= total4) return;
    const float4 g = reinterpret_cast<const float4*>(G)[i];
    const int col = (i * 4) & (HDIM - 1);
    const float4 b = *reinterpret_cast<const float4*>(bias + col);
    float4 h;
    h.x = fmaxf(g.x + b.x, 0.0f);
    h.y = fmaxf(g.y + b.y, 0.0f);
    h.z = fmaxf(g.z + b.z, 0.0f);
    h.w = fmaxf(g.w + b.w, 0.0f);
    reinterpret_cast<float4*>(H)[i] = h;
}

// ---------------- edge MLP: log_softmax([h[src],h[dst],ea] @ Wfc + bfc) ----------------
// One wave per 16-edge tile; C tile = 16 edges x 16 classes.
__global__ __launch_bounds__(256)
void k_edge_mlp_wmma(const float* __restrict__ H3, const float* __restrict__ EA,
                     const int* __restrict__ src, const int* __restrict__ dst,
                     const float* __restrict__ Wfc, const float* __restrict__ bfc,
                     float* __restrict__ out, int ntiles) {
    const int wave = threadIdx.x >> 5;
    const int lane = threadIdx.x & 31;
    const int half = lane >> 4;
    const int lrow = lane & 15;
    const int tile = blockIdx.x * 8 + wave;
    if (tile >= ntiles) return;          // wave-uniform -> EXEC all-ones inside

    const int edge = tile * 16 + lrow;   // A row owned by this lane
    const int s = src[edge], d = dst[edge];
    const float* hs = H3 + (size_t)s * HDIM + 2 * half;
    const float* hd = H3 + (size_t)d * HDIM + 2 * half;
    const float* ea = EA + (size_t)edge * 32 + 2 * half;

    v8f c = {};
    // K region 0..127 : h[src]
    #pragma unroll 4
    for (int k0 = 0; k0 < 128; k0 += 4) {
        v2f a; a.x = hs[k0]; a.y = hs[k0 + 1];
        const int kk = k0 + 2 * half;
        v2f b; b.x = Wfc[kk * 16 + lrow]; b.y = Wfc[(kk + 1) * 16 + lrow];
        c = __builtin_amdgcn_wmma_f32_16x16x4_f32(false, a, false, b,
                                                  (short)0, c, false, false);
    }
    // K region 128..255 : h[dst]
    #pragma unroll 4
    for (int k0 = 0; k0 < 128; k0 += 4) {
        v2f a; a.x = hd[k0]; a.y = hd[k0 + 1];
        const int kk = 128 + k0 + 2 * half;
        v2f b; b.x = Wfc[kk * 16 + lrow]; b.y = Wfc[(kk + 1) * 16 + lrow];
        c = __builtin_amdgcn_wmma_f32_16x16x4_f32(false, a, false, b,
                                                  (short)0, c, false, false);
    }
    // K region 256..287 : edge_attr
    #pragma unroll
    for (int k0 = 0; k0 < 32; k0 += 4) {
        v2f a; a.x = ea[k0]; a.y = ea[k0 + 1];
        const int kk = 256 + k0 + 2 * half;
        v2f b; b.x = Wfc[kk * 16 + lrow]; b.y = Wfc[(kk + 1) * 16 + lrow];
        c = __builtin_amdgcn_wmma_f32_16x16x4_f32(false, a, false, b,
                                                  (short)0, c, false, false);
    }

    const float bb = bfc[lrow];          // class bias (col = lrow in both halves)
    #pragma unroll
    for (int i = 0; i < 8; ++i) {
        float val = c[i] + bb;           // one row's 16 classes live in one 16-lane group
        float m = val;
        #pragma unroll
        for (int msk = 8; msk; msk >>= 1) m = fmaxf(m, __shfl_xor(m, msk, 16));
        float ex = expf(val - m);
        float ssum = ex;
        #pragma unroll
        for (int msk = 8; msk; msk >>= 1) ssum += __shfl_xor(ssum, msk, 16);
        const int erow = tile * 16 + i + 8 * half;
        out[(size_t)erow * 16 + lrow] = val - m - logf(ssum);
    }
}

// ---------------- launcher ----------------
extern "C" void kernel_launch(void* const* d_in, const int* in_sizes, int n_in,
                              void* d_out, int out_size, void* d_ws, size_t ws_size,
                              hipStream_t stream) {
    const float* x   = (const float*)d_in[0];
    const int*   ei  = (const int*)  d_in[1];
    const float* ea  = (const float*)d_in[2];
    const float* W1  = (const float*)d_in[3];
    const float* b1  = (const float*)d_in[4];
    const float* W2  = (const float*)d_in[5];
    const float* b2  = (const float*)d_in[6];
    const float* W3  = (const float*)d_in[7];
    const float* b3  = (const float*)d_in[8];
    const float* Wfc = (const float*)d_in[9];
    const float* bfc = (const float*)d_in[10];
    float* out = (float*)d_out;

    const int N = in_sizes[0] / HDIM;    // 100000
    const int E = in_sizes[1] / 2;       // 1000000

    // workspace layout (floats): F0 | F1 | G | dinv
    float* F0   = (float*)d_ws;
    float* F1   = F0 + (size_t)N * HDIM;
    float* G    = F1 + (size_t)N * HDIM;
    float* dinv = G  + (size_t)N * HDIM;

    const int* srcp = ei;
    const int* dstp = ei + E;

    const int TPB = 256;

    // degree -> dinv
    k_init_deg   <<<(N + TPB - 1) / TPB, TPB, 0, stream>>>(dinv, N);
    k_deg_scatter<<<(E + TPB - 1) / TPB, TPB, 0, stream>>>(dstp, dinv, E);
    k_deg_rsqrt  <<<(N + TPB - 1) / TPB, TPB, 0, stream>>>(dinv, N);

    const int gemmGrid    = N / 16;                       // 6250 (N % 16 == 0)
    const int scatterGrid = (E * 32 + TPB - 1) / TPB;     // 125000
    const int total4      = N * HDIM / 4;
    const int reluGrid    = (total4 + TPB - 1) / TPB;

    // layer 1
    k_gemm_wmma   <<<gemmGrid, TPB, 0, stream>>>(x,  W1, F0, G, dinv);
    k_edge_scatter<<<scatterGrid, TPB, 0, stream>>>(F0, srcp, dstp, dinv, G, E);
    k_bias_relu   <<<reluGrid, TPB, 0, stream>>>(G, b1, F1, total4);
    // layer 2
    k_gemm_wmma   <<<gemmGrid, TPB, 0, stream>>>(F1, W2, F0, G, dinv);
    k_edge_scatter<<<scatterGrid, TPB, 0, stream>>>(F0, srcp, dstp, dinv, G, E);
    k_bias_relu   <<<reluGrid, TPB, 0, stream>>>(G, b2, F1, total4);
    // layer 3
    k_gemm_wmma   <<<gemmGrid, TPB, 0, stream>>>(F1, W3, F0, G, dinv);
    k_edge_scatter<<<scatterGrid, TPB, 0, stream>>>(F0, srcp, dstp, dinv, G, E);
    k_bias_relu   <<<reluGrid, TPB, 0, stream>>>(G, b3, F1, total4);

    // edge MLP + log_softmax
    const int ntiles = E / 16;                            // 62500
    const int mlpGrid = (ntiles + 7) / 8;
    k_edge_mlp_wmma<<<mlpGrid, TPB, 0, stream>>>(F1, ea, srcp, dstp, Wfc, bfc,
                                                 out, ntiles);
}